// FastWeightAttention_11897059410226
// MI455X (gfx1250) — compile-verified
//
#include <hip/hip_runtime.h>
#include <math.h>

// ---------------------------------------------------------------------------
// FastWeightAttention for MI455X (gfx1250), fp32 end-to-end.
//   q = x@Wq, k = x@Wk, v = x@Wv          (WMMA f32 16x16x4 GEMMs)
//   beta = sigmoid(x@Wg)
//   v_ex = W k ; W' = W + beta(v - v_ex) k^T ; pre = W' q   (streaming kernel)
//   out = pre@Wo + bo                      (WMMA f32 GEMM)
// d_out = [ out (1024*1024) | new_weights (1024*16*64*64) ]
//
// GEMM uses GLOBAL_LOAD_ASYNC_TO_LDS_B128 (ASYNCcnt) with double-buffered LDS,
// 128x128 block tiles, 32x64 wave tiles (8 f32-WMMA accumulators / wave).
// ---------------------------------------------------------------------------

typedef float v2f __attribute__((ext_vector_type(2)));
typedef float v8f __attribute__((ext_vector_type(8)));

#define MB  1024      // batch
#define DM  1024      // d_model
#define NH  16        // heads
#define DK  64        // d_k

#define AP  20        // A LDS row stride (floats): 80B, 16B aligned, bank-clean
#define BP  140       // B LDS row stride (floats): 560B, 16B aligned, bank-clean

// per-lane 16B global -> LDS async copy (no VGPR staging, ASYNCcnt tracked)
__device__ __forceinline__ void async_copy_b128(unsigned lds_byte_off,
                                                const float* gaddr) {
    asm volatile("global_load_async_to_lds_b128 %0, %1, off"
                 :: "v"(lds_byte_off), "v"(gaddr) : "memory");
}
__device__ __forceinline__ unsigned lds_off(const void* p) {
    // flat LDS address: low 32 bits are the LDS byte offset (ISA 10.2)
    return (unsigned)(unsigned long long)p;
}

// ---------------- WMMA fp32 GEMM: C[M,N] = A[M,K] @ B[K,N] (+bias) ---------
// Block: 256 threads (8 waves). Block tile 128x128; wave tile 32x64
// (2 M-tiles x 4 N-tiles of 16x16). K staged through LDS in slices of 16,
// double-buffered via async-to-LDS copies.
__global__ __launch_bounds__(256)
void gemm_wmma_f32(const float* __restrict__ A, const float* __restrict__ B,
                   const float* __restrict__ bias, float* __restrict__ C,
                   int M, int N, int K)
{
    __shared__ __align__(16) float As[2][128][AP];  // row-major A tile [m][k]
    __shared__ __align__(16) float Bs[2][16][BP];   // row-major B tile [k][n]

    const int t    = threadIdx.x;
    const int wave = t >> 5;
    const int lane = t & 31;
    const int hf   = lane >> 4;      // 16-lane half: selects K+0/1 vs K+2/3
    const int l16  = lane & 15;
    const int wm   = wave >> 1;      // 0..3 -> rows wm*32
    const int wn   = wave & 1;       // 0..1 -> cols wn*64
    const int bm   = blockIdx.y * 128;
    const int bn   = blockIdx.x * 128;

    // issue one K-slice (16 wide) of async copies: 4 x b128 per thread
    auto issue = [&](int kt, int b) {
        #pragma unroll
        for (int i = 0; i < 2; ++i) {
            const int idx = t + (i << 8);              // 0..511
            // A tile: 128 rows x 16 k
            const int am = idx >> 2;                   // 0..127
            const int ak = (idx & 3) << 2;             // 0,4,8,12
            async_copy_b128(lds_off(&As[b][am][ak]),
                            &A[(size_t)(bm + am) * K + kt + ak]);
            // B tile: 16 rows x 128 n
            const int br = idx >> 5;                   // 0..15
            const int bc = (idx & 31) << 2;            // 0..124
            async_copy_b128(lds_off(&Bs[b][br][bc]),
                            &B[(size_t)(kt + br) * N + bn + bc]);
        }
    };

    v8f acc[2][4] = {};

    issue(0, 0);
    const int nslice = K >> 4;
    for (int s = 0; s < nslice; ++s) {
        const int b = s & 1;
        if (s + 1 < nslice) {
            issue((s + 1) << 4, b ^ 1);
            asm volatile("s_wait_asynccnt 0x4" ::: "memory");  // cur slice done
        } else {
            asm volatile("s_wait_asynccnt 0x0" ::: "memory");
        }
        __syncthreads();

        #pragma unroll
        for (int kk = 0; kk < 4; ++kk) {
            const int k0 = kk * 4 + 2 * hf;
            // A frags: contiguous b64 per lane (rows padded to AP floats)
            const v2f a0 = *reinterpret_cast<const v2f*>(
                &As[b][wm * 32 + l16][k0]);
            const v2f a1 = *reinterpret_cast<const v2f*>(
                &As[b][wm * 32 + 16 + l16][k0]);
            #pragma unroll
            for (int nt = 0; nt < 4; ++nt) {
                v2f bf;
                bf.x = Bs[b][k0    ][wn * 64 + nt * 16 + l16];
                bf.y = Bs[b][k0 + 1][wn * 64 + nt * 16 + l16];
                acc[0][nt] = __builtin_amdgcn_wmma_f32_16x16x4_f32(
                    false, a0, false, bf, (short)0, acc[0][nt], false, false);
                acc[1][nt] = __builtin_amdgcn_wmma_f32_16x16x4_f32(
                    false, a1, false, bf, (short)0, acc[1][nt], false, false);
            }
        }
        __syncthreads();
    }

    // store: VGPR r holds M = r (lanes 0-15) / M = r+8 (lanes 16-31)
    #pragma unroll
    for (int mt = 0; mt < 2; ++mt) {
        #pragma unroll
        for (int nt = 0; nt < 4; ++nt) {
            const int col = bn + wn * 64 + nt * 16 + l16;
            const float bb = bias ? bias[col] : 0.0f;
            #pragma unroll
            for (int r = 0; r < 8; ++r) {
                const int row = bm + wm * 32 + mt * 16 + hf * 8 + r;
                C[(size_t)row * N + col] = acc[mt][nt][r] + bb;
            }
        }
    }
}

// ---------------- gate: beta[b,h] = sigmoid(sum_d x[b,d] * Wg[d,h]) --------
__global__ __launch_bounds__(256)
void gate_sigmoid(const float* __restrict__ x, const float* __restrict__ Wg,
                  float* __restrict__ beta)
{
    __shared__ float red[256];
    const int b = blockIdx.x;
    const int t = threadIdx.x;
    const int h = t & 15;
    const int chunk = t >> 4;           // 16 chunks of 64 along d_model
    float s = 0.f;
    #pragma unroll 4
    for (int i = 0; i < 64; ++i) {
        const int d = chunk * 64 + i;
        s += x[b * DM + d] * Wg[d * NH + h];
    }
    red[t] = s;
    __syncthreads();
    if (t < NH) {
        float acc = 0.f;
        #pragma unroll
        for (int c = 0; c < 16; ++c) acc += red[c * 16 + t];
        beta[b * NH + t] = 1.0f / (1.0f + __expf(-acc));
    }
}

// ---------------- fast-weight delta-rule update (streaming, BW-bound) ------
// One block per (b,h). Thread t owns W[r][cq*16 .. cq*16+15], r = t/4, cq = t%4.
__global__ __launch_bounds__(256)
void fw_update(const float* __restrict__ weights,
               const float* __restrict__ qb, const float* __restrict__ kb,
               const float* __restrict__ vb, const float* __restrict__ beta,
               float* __restrict__ new_weights, float* __restrict__ out_pre)
{
    __shared__ __align__(16) float kv[DK], qv[DK], vv[DK], dl[DK];
    __shared__ float red[256];
    __shared__ float bet;

    const int bh = blockIdx.x;              // b*16 + h
    const int t  = threadIdx.x;
    const int r  = t >> 2;                  // 0..63 (v index / row)
    const int cq = t & 3;                   // 16-col quarter
    const size_t wbase = (size_t)bh * (DK * DK);
    const int vecbase  = bh * DK;           // q/k/v flat: [b*1024 + h*64 + i]

    if      (t < 64)   kv[t]        = kb[vecbase + t];
    else if (t < 128)  qv[t - 64]   = qb[vecbase + (t - 64)];
    else if (t < 192)  vv[t - 128]  = vb[vecbase + (t - 128)];
    else if (t == 192) bet          = beta[bh];
    __syncthreads();

    // load owned 16-float slice of W (4 x float4, coalesced)
    float4 w[4];
    const float4* wrow =
        reinterpret_cast<const float4*>(weights + wbase + r * DK + cq * 16);
    #pragma unroll
    for (int i = 0; i < 4; ++i) w[i] = wrow[i];

    // v_existing = W k (partial per thread, LDS reduce 4 -> 1 per row)
    float part = 0.f;
    #pragma unroll
    for (int i = 0; i < 4; ++i) {
        const float4 k4 = reinterpret_cast<const float4*>(&kv[cq * 16])[i];
        part += w[i].x * k4.x + w[i].y * k4.y + w[i].z * k4.z + w[i].w * k4.w;
    }
    red[t] = part;
    __syncthreads();
    if (t < 64) {
        const float vex = red[4 * t] + red[4 * t + 1] + red[4 * t + 2] + red[4 * t + 3];
        dl[t] = bet * (vv[t] - vex);
    }
    __syncthreads();

    // W' = W + dl[r] * k^T  -> stream out; pre[r] = W' q
    const float d = dl[r];
    float4* orow =
        reinterpret_cast<float4*>(new_weights + wbase + r * DK + cq * 16);
    float part2 = 0.f;
    #pragma unroll
    for (int i = 0; i < 4; ++i) {
        const float4 k4 = reinterpret_cast<const float4*>(&kv[cq * 16])[i];
        const float4 q4 = reinterpret_cast<const float4*>(&qv[cq * 16])[i];
        float4 wn;
        wn.x = w[i].x + d * k4.x;
        wn.y = w[i].y + d * k4.y;
        wn.z = w[i].z + d * k4.z;
        wn.w = w[i].w + d * k4.w;
        orow[i] = wn;
        part2 += wn.x * q4.x + wn.y * q4.y + wn.z * q4.z + wn.w * q4.w;
    }
    __syncthreads();                         // red reuse
    red[t] = part2;
    __syncthreads();
    if (t < 64) {
        out_pre[vecbase + t] =
            red[4 * t] + red[4 * t + 1] + red[4 * t + 2] + red[4 * t + 3];
    }
}

// ---------------------------------------------------------------------------
extern "C" void kernel_launch(void* const* d_in, const int* in_sizes, int n_in,
                              void* d_out, int out_size, void* d_ws, size_t ws_size,
                              hipStream_t stream)
{
    const float* x       = (const float*)d_in[0];
    const float* weights = (const float*)d_in[1];
    const float* Wq      = (const float*)d_in[2];
    const float* Wk      = (const float*)d_in[3];
    const float* Wv      = (const float*)d_in[4];
    const float* Wg      = (const float*)d_in[5];
    const float* Wo      = (const float*)d_in[6];
    const float* bo      = (const float*)d_in[7];

    float* out   = (float*)d_out;                       // [1024*1024]
    float* new_w = out + (size_t)MB * DM;               // [1024*16*64*64]

    float* ws   = (float*)d_ws;
    float* q    = ws;
    float* k    = ws + (1u << 20);
    float* v    = ws + (2u << 20);
    float* beta = ws + (3u << 20);                      // 16384 used
    float* pre  = beta + 16384;                         // 1M floats

    const dim3 gg(DM / 128, MB / 128);                  // 8 x 8 blocks
    const dim3 gb(256);

    gemm_wmma_f32<<<gg, gb, 0, stream>>>(x, Wq, nullptr, q, MB, DM, DM);
    gemm_wmma_f32<<<gg, gb, 0, stream>>>(x, Wk, nullptr, k, MB, DM, DM);
    gemm_wmma_f32<<<gg, gb, 0, stream>>>(x, Wv, nullptr, v, MB, DM, DM);
    gate_sigmoid<<<MB, 256, 0, stream>>>(x, Wg, beta);
    fw_update<<<MB * NH, 256, 0, stream>>>(weights, q, k, v, beta, new_w, pre);
    gemm_wmma_f32<<<gg, gb, 0, stream>>>(pre, Wo, bo, out, MB, DM, DM);
}